// UniqueEq3Net_75900662055583
// MI455X (gfx1250) — compile-verified
//
#include <hip/hip_runtime.h>

typedef __attribute__((ext_vector_type(2))) float v2f;
typedef __attribute__((ext_vector_type(8))) float v8f;

// Problem constants: B=8, n=32, IN=16, HID=D=O=32, OUT=32, P=12.
// y[b,o,i,j,k] = sum_d [ c0 v_i v_j v_k + c1 S v_j v_k + c2 S v_i v_k + c3 S v_i v_j
//                      + c4 S^2 v_k + c5 S^2 v_j + c6 S^2 v_i + c7 S^3
//                      + c8 v_i^3 + c9 v_j^3 + c10 v_k^3 + c11 C3 ] + eqb[o]
// with v = V[d,:] = h[b,:,d], S = sum_i v_i, C3 = sum_i v_i^3.

// ---------------- Kernel 1: per-batch precompute ----------------
__global__ __launch_bounds__(256) void eq3_precompute(
    const float* __restrict__ x,  const float* __restrict__ W1, const float* __restrict__ b1,
    const float* __restrict__ W2, const float* __restrict__ b2,
    const float* __restrict__ eq, const float* __restrict__ eqb,
    float* __restrict__ wsV, float* __restrict__ wsA, float* __restrict__ wsPj,
    float* __restrict__ wsPk, float* __restrict__ wsC)
{
    __shared__ float xs[32 * 16];
    __shared__ float h1s[32 * 32];
    __shared__ float Vs[32 * 32];      // V[d][i]
    __shared__ float Ss[32];
    __shared__ float C3s[32];
    __shared__ float w2s[1024];        // c2[d,o]*S[d]
    __shared__ float w3s[1024];        // c3[d,o]*S[d]
    __shared__ float Qjs[1024];        // i-independent part of Pj: [o][j]
    __shared__ float Qks[1024];        // i-independent part of Pk: [o][k]

    const int b = blockIdx.x;
    const int tid = threadIdx.x;

    for (int idx = tid; idx < 512; idx += 256) xs[idx] = x[b * 512 + idx];
    __syncthreads();

    // h1 = relu(x @ W1 + b1)   (32 x 32)
    for (int idx = tid; idx < 1024; idx += 256) {
        int i = idx >> 5, m = idx & 31;
        float s = b1[m];
        for (int t = 0; t < 16; ++t) s += xs[i * 16 + t] * W1[t * 32 + m];
        h1s[idx] = fmaxf(s, 0.f);
    }
    __syncthreads();

    // V[d][i] = relu(h1 @ W2 + b2)[i][d]
    for (int idx = tid; idx < 1024; idx += 256) {
        int i = idx >> 5, d = idx & 31;
        float s = b2[d];
        for (int m = 0; m < 32; ++m) s += h1s[i * 32 + m] * W2[m * 32 + d];
        float v = fmaxf(s, 0.f);
        Vs[d * 32 + i] = v;
        wsV[b * 1024 + d * 32 + i] = v;
    }
    __syncthreads();

    if (tid < 32) {
        int d = tid; float s = 0.f, c3 = 0.f;
        for (int i = 0; i < 32; ++i) { float v = Vs[d * 32 + i]; s += v; c3 += v * v * v; }
        Ss[d] = s; C3s[d] = c3;
    }
    __syncthreads();

    for (int idx = tid; idx < 1024; idx += 256) {
        int d = idx >> 5, o = idx & 31;
        int e = (d * 32 + o) * 12;
        w3s[d * 32 + o] = eq[e + 3] * Ss[d];
        w2s[d * 32 + o] = eq[e + 2] * Ss[d];
    }
    __syncthreads();

    // Qj[o][j] = sum_d c5 S^2 v_j + c9 v_j^3 ; Qk[o][k] = sum_d c4 S^2 v_k + c10 v_k^3
    for (int idx = tid; idx < 1024; idx += 256) {
        int o = idx >> 5, jj = idx & 31;
        float qj = 0.f, qk = 0.f;
        for (int d = 0; d < 32; ++d) {
            int e = (d * 32 + o) * 12;
            float vj = Vs[d * 32 + jj];
            float S2 = Ss[d] * Ss[d];
            float vj3 = vj * vj * vj;
            qj += eq[e + 5] * S2 * vj + eq[e + 9]  * vj3;
            qk += eq[e + 4] * S2 * vj + eq[e + 10] * vj3;
        }
        Qjs[idx] = qj; Qks[idx] = qk;
    }
    __syncthreads();

    // A[(o*32+i)][d] = c0[d,o] v[d,i] + c1[d,o] S[d]   (K side of the big GEMM)
    for (int idx = tid; idx < 32768; idx += 256) {
        int r = idx >> 5, d = idx & 31;
        int o = r >> 5, i = r & 31;
        int e = (d * 32 + o) * 12;
        wsA[b * 32768 + idx] = eq[e + 0] * Vs[d * 32 + i] + eq[e + 1] * Ss[d];
    }

    // Pj[(o,i)][j] and Pk[(o,i)][k]
    for (int idx = tid; idx < 32768; idx += 256) {
        int r = idx >> 5, jj = idx & 31;
        int o = r >> 5, i = r & 31;
        float sj = Qjs[o * 32 + jj], sk = Qks[o * 32 + jj];
        for (int d = 0; d < 32; ++d) {
            float p = Vs[d * 32 + i] * Vs[d * 32 + jj];
            sj += w3s[d * 32 + o] * p;
            sk += w2s[d * 32 + o] * p;
        }
        wsPj[b * 32768 + idx] = sj;
        wsPk[b * 32768 + idx] = sk;
    }

    // Cst[(o,i)] = sum_d c6 S^2 v_i + c7 S^3 + c8 v_i^3 + c11 C3  + eqb[o]
    for (int idx = tid; idx < 1024; idx += 256) {
        int o = idx >> 5, i = idx & 31;
        float s = eqb[o];
        for (int d = 0; d < 32; ++d) {
            int e = (d * 32 + o) * 12;
            float S1 = Ss[d], S2 = S1 * S1;
            float vi = Vs[d * 32 + i];
            s += eq[e + 6] * S2 * vi + eq[e + 7] * S2 * S1
               + eq[e + 8] * vi * vi * vi + eq[e + 11] * C3s[d];
        }
        wsC[b * 1024 + idx] = s;
    }
}

// ---------------- Kernel 2: fused WMMA GEMM + relu + fc GEMM + reduce ----------------
// grid: 8 b x 64 jk-blocks. Block owns one j and 16 consecutive k columns.
__global__ __launch_bounds__(256) void eq3_main(
    const float* __restrict__ wsV, const float* __restrict__ wsA,
    const float* __restrict__ wsPj, const float* __restrict__ wsPk,
    const float* __restrict__ wsC, const float* __restrict__ fcw,
    const float* __restrict__ fcb, float* __restrict__ part)
{
    __shared__ float Vs[1024];
    __shared__ float Y[32 * 512];   // relu_y as [o][i*16 + col]
    __shared__ float facc[256];     // per-wave row partials [wave][m]

    const int gid = blockIdx.x;
    const int b = gid >> 6, blk = gid & 63;
    const int j = blk >> 1, k0 = (blk & 1) << 4;
    const int tid = threadIdx.x;
    const int wave = tid >> 5, lane = tid & 31;
    const int hi = lane >> 4, ln = lane & 15;

    for (int idx = tid; idx < 1024; idx += 256) Vs[idx] = wsV[b * 1024 + idx];
    facc[tid] = 0.f;
    __syncthreads();

    // B fragments for phase A: B[d, col] = v[d,j] * v[d,k0+col]
    // f32 4x16 B layout: vgpr0 = K = 4kk+2*hi, vgpr1 = K = 4kk+2*hi+1, N = ln
    v2f bf[8];
#pragma unroll
    for (int kk = 0; kk < 8; ++kk) {
        int d0 = 4 * kk + 2 * hi;
        bf[kk].x = Vs[d0 * 32 + j]       * Vs[d0 * 32 + k0 + ln];
        bf[kk].y = Vs[(d0 + 1) * 32 + j] * Vs[(d0 + 1) * 32 + k0 + ln];
    }

    const float* Ab  = wsA  + b * 32768;
    const float* Pjb = wsPj + b * 32768;
    const float* Pkb = wsPk + b * 32768;
    const float* Cb  = wsC  + b * 1024;

    // Phase A: T0[(o,i), col] tiles; wave w covers M-tiles 8w..8w+7 (rows 0..1023)
    for (int t = 0; t < 8; ++t) {
        int rb = (wave * 8 + t) * 16;
        int r = rb + ln;                       // A-frag row (lane layout: M = ln)
        v8f c = {0.f, 0.f, 0.f, 0.f, 0.f, 0.f, 0.f, 0.f};
#pragma unroll
        for (int kk = 0; kk < 8; ++kk) {
            v2f a = *(const v2f*)(Ab + r * 32 + 4 * kk + 2 * hi);
            c = __builtin_amdgcn_wmma_f32_16x16x4_f32(false, a, false, bf[kk],
                                                      (short)0, c, false, false);
        }
        // epilogue: separable terms + relu, stage into Y (C/D layout: M = v + 8*hi)
#pragma unroll
        for (int v = 0; v < 8; ++v) {
            int rv = rb + v + 8 * hi;
            float y = c[v] + Pjb[rv * 32 + j] + Pkb[rv * 32 + k0 + ln] + Cb[rv];
            y = fmaxf(y, 0.f);
            Y[(rv >> 5) * 512 + (rv & 31) * 16 + ln] = y;
        }
    }
    __syncthreads();

    // Phase C: z[m, q] = sum_o fcw[o,m] * Y[o,q], q = i*16+col (N=512), K=o=32
#pragma unroll
    for (int mt = 0; mt < 2; ++mt) {
        for (int u = 0; u < 4; ++u) {
            int q = (wave * 4 + u) * 16 + ln;
            int m = mt * 16 + ln;
            v8f c = {0.f, 0.f, 0.f, 0.f, 0.f, 0.f, 0.f, 0.f};
#pragma unroll
            for (int kk = 0; kk < 8; ++kk) {
                int d0 = 4 * kk + 2 * hi;
                v2f a;  a.x = fcw[d0 * 32 + m];       a.y = fcw[(d0 + 1) * 32 + m];
                v2f bb; bb.x = Y[d0 * 512 + q];       bb.y = Y[(d0 + 1) * 512 + q];
                c = __builtin_amdgcn_wmma_f32_16x16x4_f32(false, a, false, bb,
                                                          (short)0, c, false, false);
            }
#pragma unroll
            for (int v = 0; v < 8; ++v) {
                int mr = mt * 16 + v + 8 * hi;
                float val = fmaxf(c[v] + fcb[mr], 0.f);
                // sum 16 columns within each lane-half
                val += __shfl_xor(val, 1);
                val += __shfl_xor(val, 2);
                val += __shfl_xor(val, 4);
                val += __shfl_xor(val, 8);
                if (ln == 0) facc[wave * 32 + mr] += val;  // lane-exclusive slot, no race
            }
        }
    }
    __syncthreads();

    if (tid < 32) {  // deterministic per-WG partial (fixed summation order)
        float s = 0.f;
        for (int w = 0; w < 8; ++w) s += facc[w * 32 + tid];
        part[gid * 32 + tid] = s;
    }
}

// ---------------- Kernel 3: reduce partials + head ----------------
__global__ __launch_bounds__(256) void eq3_final(
    const float* __restrict__ part, const float* __restrict__ outw,
    const float* __restrict__ outb, float* __restrict__ out)
{
    int tid = threadIdx.x;
    int b = tid >> 5, m = tid & 31;
    float s = 0.f;
    for (int blk = 0; blk < 64; ++blk) s += part[(b * 64 + blk) * 32 + m];
    float val = fmaxf(s, 0.f) * outw[m];
    val += __shfl_xor(val, 1);
    val += __shfl_xor(val, 2);
    val += __shfl_xor(val, 4);
    val += __shfl_xor(val, 8);
    val += __shfl_xor(val, 16);
    if (m == 0) out[b] = val + outb[0];
}

extern "C" void kernel_launch(void* const* d_in, const int* in_sizes, int n_in,
                              void* d_out, int out_size, void* d_ws, size_t ws_size,
                              hipStream_t stream) {
    const float* x   = (const float*)d_in[0];
    const float* W1  = (const float*)d_in[1];
    const float* b1  = (const float*)d_in[2];
    const float* W2  = (const float*)d_in[3];
    const float* b2  = (const float*)d_in[4];
    const float* eq  = (const float*)d_in[5];
    const float* eqb = (const float*)d_in[6];
    const float* fcw = (const float*)d_in[7];
    const float* fcb = (const float*)d_in[8];
    const float* ow  = (const float*)d_in[9];
    const float* ob  = (const float*)d_in[10];
    (void)in_sizes; (void)n_in; (void)out_size; (void)ws_size;

    float* ws   = (float*)d_ws;
    float* wsV  = ws;                        // 8 * 1024
    float* wsA  = wsV  + 8 * 1024;           // 8 * 32768
    float* wsPj = wsA  + 8 * 32768;          // 8 * 32768
    float* wsPk = wsPj + 8 * 32768;          // 8 * 32768
    float* wsC  = wsPk + 8 * 32768;          // 8 * 1024
    float* part = wsC  + 8 * 1024;           // 512 * 32
    // total: 819,200 floats = ~3.3 MB

    eq3_precompute<<<8, 256, 0, stream>>>(x, W1, b1, W2, b2, eq, eqb,
                                          wsV, wsA, wsPj, wsPk, wsC);
    eq3_main<<<512, 256, 0, stream>>>(wsV, wsA, wsPj, wsPk, wsC, fcw, fcb, part);
    eq3_final<<<1, 256, 0, stream>>>(part, ow, ob, (float*)d_out);
}